// QuantumMultiHeadAttention_42786464203440
// MI455X (gfx1250) — compile-verified
//
#include <hip/hip_runtime.h>
#include <hip/hip_bf16.h>

// ---------------------------------------------------------------------------
// Quantum multi-head attention, fused for MI455X (gfx1250, wave32).
//   NQ=4, DIM=16, B=32, S=2048, E=4.
// Kernel 1: per-sample 4-qubit statevector sim -> Q,K,V (shared embedding).
// Kernel 2: flash attention with V_WMMA_F32_16X16X4_F32 score tiles
//           (key-major D layout => purely in-lane softmax per half-wave;
//           single two-way softmax merge at the end), then the output
//           circuit applied in-register.
// ---------------------------------------------------------------------------

typedef __attribute__((ext_vector_type(2))) float v2f;
typedef __attribute__((ext_vector_type(8))) float v8f;

#define NQ   4
#define DIM  16
#define BATCH 32
#define SEQ  2048
#define NSAMP (BATCH * SEQ)

// RX(theta) on the qubit whose index bit mask is MASK (mask = 1<<(3-q)).
// a0' = c*a0 - i*s*a1 ; a1' = c*a1 - i*s*a0   (c=cos(t/2), s=sin(t/2))
__device__ __forceinline__ void rx_apply(float* re, float* im,
                                         float half_theta, int mask) {
  float c = __cosf(half_theta);
  float s = __sinf(half_theta);
#pragma unroll
  for (int i = 0; i < DIM; ++i) {
    if (i & mask) continue;
    const int j = i | mask;
    float a0r = re[i], a0i = im[i], a1r = re[j], a1i = im[j];
    re[i] = c * a0r + s * a1i;
    im[i] = c * a0i - s * a1r;
    re[j] = c * a1r + s * a0i;
    im[j] = c * a1i - s * a0r;
  }
}

// CNOT ring: for q=0..3, control q, target (q+1)%4; new[i]=old[perm[i]].
__device__ __forceinline__ void cnot_ring(float* re, float* im) {
#pragma unroll
  for (int q = 0; q < NQ; ++q) {
    const int cm = 1 << (3 - q);
    const int tm = 1 << (3 - ((q + 1) & 3));
    float tr[DIM], ti[DIM];
#pragma unroll
    for (int i = 0; i < DIM; ++i) {
      const int src = (i & cm) ? (i ^ tm) : i;
      tr[i] = re[src];
      ti[i] = im[src];
    }
#pragma unroll
    for (int i = 0; i < DIM; ++i) { re[i] = tr[i]; im[i] = ti[i]; }
  }
}

// <Z_q> = sum_i |amp_i|^2 * (1 - 2*bit_q(i))
__device__ __forceinline__ void expvals(const float* re, const float* im,
                                        float* out) {
  float p[DIM];
#pragma unroll
  for (int i = 0; i < DIM; ++i) p[i] = re[i] * re[i] + im[i] * im[i];
#pragma unroll
  for (int q = 0; q < NQ; ++q) {
    const int m = 1 << (3 - q);
    float acc = 0.f;
#pragma unroll
    for (int i = 0; i < DIM; ++i) acc += (i & m) ? -p[i] : p[i];
    out[q] = acc;
  }
}

// Entangler layer (RX(w) ring + CNOT) + measurement, from a given state.
__device__ __forceinline__ void layer_measure(const float* re0, const float* im0,
                                              const float* w, float* out) {
  float re[DIM], im[DIM];
#pragma unroll
  for (int i = 0; i < DIM; ++i) { re[i] = re0[i]; im[i] = im0[i]; }
  rx_apply(re, im, 0.5f * w[0], 8);
  rx_apply(re, im, 0.5f * w[1], 4);
  rx_apply(re, im, 0.5f * w[2], 2);
  rx_apply(re, im, 0.5f * w[3], 1);
  cnot_ring(re, im);
  expvals(re, im, out);
}

// Full circuit from |0>: embedding RX(angles) then layer + measure.
__device__ __forceinline__ void full_circuit(const float* ang, const float* w,
                                             float* out) {
  float re[DIM], im[DIM];
#pragma unroll
  for (int i = 0; i < DIM; ++i) { re[i] = 0.f; im[i] = 0.f; }
  re[0] = 1.f;
  rx_apply(re, im, 0.5f * ang[0], 8);
  rx_apply(re, im, 0.5f * ang[1], 4);
  rx_apply(re, im, 0.5f * ang[2], 2);
  rx_apply(re, im, 0.5f * ang[3], 1);
  layer_measure(re, im, w, out);
}

// ---------------------------------------------------------------------------
// Kernel 1: Q,K,V circuits. Embedding state shared across the three heads.
// ---------------------------------------------------------------------------
__global__ void qkv_kernel(const float* __restrict__ x,
                           const float* __restrict__ wq,
                           const float* __restrict__ wk,
                           const float* __restrict__ wv,
                           float* __restrict__ Q,
                           float* __restrict__ K,
                           float* __restrict__ V) {
  const int i = blockIdx.x * blockDim.x + threadIdx.x;  // exact grid
  float ang[NQ];
#pragma unroll
  for (int e = 0; e < NQ; ++e) ang[e] = x[(size_t)i * NQ + e];

  float re[DIM], im[DIM];
#pragma unroll
  for (int k = 0; k < DIM; ++k) { re[k] = 0.f; im[k] = 0.f; }
  re[0] = 1.f;
  rx_apply(re, im, 0.5f * ang[0], 8);
  rx_apply(re, im, 0.5f * ang[1], 4);
  rx_apply(re, im, 0.5f * ang[2], 2);
  rx_apply(re, im, 0.5f * ang[3], 1);

  float o[NQ];
  layer_measure(re, im, wq, o);
#pragma unroll
  for (int e = 0; e < NQ; ++e) Q[(size_t)i * NQ + e] = o[e];
  layer_measure(re, im, wk, o);
#pragma unroll
  for (int e = 0; e < NQ; ++e) K[(size_t)i * NQ + e] = o[e];
  layer_measure(re, im, wv, o);
#pragma unroll
  for (int e = 0; e < NQ; ++e) V[(size_t)i * NQ + e] = o[e];
}

// ---------------------------------------------------------------------------
// Kernel 2: flash attention (one wave per 16-query tile) + output circuit.
//   Score tile computed key-major: D = K_tile(16x4) x Q_tile^T(4x16) via
//   v_wmma_f32_16x16x4_f32.  D layout: VGPR r, lane L<16 -> (key=r, q=L);
//   lane>=16 -> (key=r+8, q=L-16).  Each half-wave runs an independent
//   online softmax over its 8-key groups (no cross-lane traffic in the
//   loop); the two partial softmaxes are merged once at the end.
// ---------------------------------------------------------------------------
__global__ void attn_kernel(const float* __restrict__ Q,
                            const float* __restrict__ K,
                            const float* __restrict__ V,
                            const float* __restrict__ wc,
                            float* __restrict__ out) {
  const int tid   = blockIdx.x * blockDim.x + threadIdx.x;
  const int wave  = tid >> 5;        // 4096 waves total
  const int lane  = threadIdx.x & 31;
  const int batch = wave >> 7;       // 128 query tiles per batch
  const int qtile = wave & 127;
  const int row   = lane & 15;       // matrix row handled by this lane
  const int hi    = lane >> 4;       // half-wave (element pair / key group)

  const float* Qb = Q + (size_t)batch * SEQ * NQ;
  const float* Kb = K + (size_t)batch * SEQ * NQ;
  const float* Vb = V + (size_t)batch * SEQ * NQ;

  // B operand: Q tile, scale 1/sqrt(E)=0.5 folded in.
  const float* qp = Qb + (size_t)(qtile * 16 + row) * NQ + hi * 2;
  v2f bq;
  bq.x = qp[0] * 0.5f;
  bq.y = qp[1] * 0.5f;

  // Per-half-wave running softmax state (merged once at the end).
  float m = -__builtin_inff();
  float l = 0.f;
  float acc0 = 0.f, acc1 = 0.f, acc2 = 0.f, acc3 = 0.f;

  for (int kt = 0; kt < SEQ / 16; ++kt) {
    // A operand: K tile (same lane addressing pattern as B).
    const float* kp = Kb + (size_t)(kt * 16 + row) * NQ + hi * 2;
    v2f a;
    a.x = kp[0];
    a.y = kp[1];

    v8f d = {};
    d = __builtin_amdgcn_wmma_f32_16x16x4_f32(
        /*neg_a=*/false, a, /*neg_b=*/false, bq,
        /*c_mod=*/(short)0, d, /*reuse_a=*/false, /*reuse_b=*/false);

    // In-lane tile max over this half's 8 keys (no cross-lane traffic).
    float tmax = fmaxf(d[0], d[1]);
#pragma unroll
    for (int r = 2; r < 8; ++r) tmax = fmaxf(tmax, d[r]);

    const float mnew  = fmaxf(m, tmax);
    const float scale = __expf(m - mnew);
    l *= scale;
    acc0 *= scale; acc1 *= scale; acc2 *= scale; acc3 *= scale;

    const int kbase = kt * 16 + hi * 8;
#pragma unroll
    for (int r = 0; r < 8; ++r) {
      const float p = __expf(d[r] - mnew);
      l += p;
      const float4 v = *reinterpret_cast<const float4*>(Vb + (size_t)(kbase + r) * NQ);
      acc0 += p * v.x;
      acc1 += p * v.y;
      acc2 += p * v.z;
      acc3 += p * v.w;
    }
    m = mnew;
  }

  // Two-way softmax merge between the half-waves (one shuffle batch total).
  const float mo = __shfl_xor(m, 16, 32);
  const float lo = __shfl_xor(l, 16, 32);
  const float b0 = __shfl_xor(acc0, 16, 32);
  const float b1 = __shfl_xor(acc1, 16, 32);
  const float b2 = __shfl_xor(acc2, 16, 32);
  const float b3 = __shfl_xor(acc3, 16, 32);

  const float mf = fmaxf(m, mo);
  const float sa = __expf(m - mf);   // my scale
  const float sb = __expf(mo - mf);  // partner scale

  const float lf = l * sa + lo * sb;
  acc0 = acc0 * sa + b0 * sb;
  acc1 = acc1 * sa + b1 * sb;
  acc2 = acc2 * sa + b2 * sb;
  acc3 = acc3 * sa + b3 * sb;

  const float inv = 1.f / lf;
  float ctx[NQ] = {acc0 * inv, acc1 * inv, acc2 * inv, acc3 * inv};

  // Output circuit on the context (both halves compute identically; keep
  // EXEC full, only the low half stores).
  float w[NQ] = {wc[0], wc[1], wc[2], wc[3]};
  float o[NQ];
  full_circuit(ctx, w, o);

  if (hi == 0) {
    const size_t q = (size_t)batch * SEQ + qtile * 16 + row;
#pragma unroll
    for (int e = 0; e < NQ; ++e) out[q * NQ + e] = o[e];
  }
}

// ---------------------------------------------------------------------------
extern "C" void kernel_launch(void* const* d_in, const int* in_sizes, int n_in,
                              void* d_out, int out_size, void* d_ws, size_t ws_size,
                              hipStream_t stream) {
  const float* x  = (const float*)d_in[0];
  const float* wq = (const float*)d_in[1];
  const float* wk = (const float*)d_in[2];
  const float* wv = (const float*)d_in[3];
  const float* wc = (const float*)d_in[4];
  float* out = (float*)d_out;

  float* ws = (float*)d_ws;                 // 3 MB used of workspace
  float* Qw = ws;
  float* Kw = ws + (size_t)NSAMP * NQ;
  float* Vw = ws + (size_t)2 * NSAMP * NQ;

  // Kernel 1: one thread per sample, exact grid (65536 threads).
  qkv_kernel<<<NSAMP / 256, 256, 0, stream>>>(x, wq, wk, wv, Qw, Kw, Vw);

  // Kernel 2: one wave per 16-query tile: 32*128 = 4096 waves = 131072 thr.
  attn_kernel<<<(BATCH * (SEQ / 16) * 32) / 256, 256, 0, stream>>>(
      Qw, Kw, Vw, wc, out);
}